// MultiWaveletTransform_29171417874965
// MI455X (gfx1250) — compile-verified
//
#include <hip/hip_runtime.h>
#include <math.h>

#ifndef M_PI
#define M_PI 3.14159265358979323846
#endif

typedef __attribute__((ext_vector_type(2))) float v2f;
typedef __attribute__((ext_vector_type(8))) float v8f;

#define NUM_WAVELETS 4
#define BATCH 64
#define SIG_LEN 131072
#define LEVELS 6
#define T_OUT 131166   // 2063+2063+4111+8207+16398+32780+65544
#define WPB 8          // waves (tiles) per block: 256 threads

// ws layout (in floats):
//   [0,128)      constrained filters, filt[f=0..7][tap=0..15]
//   [256, 6400)  Toeplitz A fragments in per-lane WMMA layout:
//                toep[((n*2+s)*12+q)*32+lane] as float2 (6144 floats)
//   [8192, ...)  ping-pong approx buffers
#define WS_TOEP 256
#define WS_BUF 8192

// ---------------------------------------------------------------------------
// Phase 1a: frequency-constrain the 8 filters (4 low + 4 high).
// Direct real DFT (size-64) in double; matches pad->rfft->(*w)->irfft->[:16],
// then L2-normalize * sqrt(2). 128 threads = 8 filters x 16 taps.
// ---------------------------------------------------------------------------
__global__ void constrain_filters_kernel(const float* __restrict__ lowf,
                                         const float* __restrict__ highf,
                                         float* __restrict__ filt_out) {
  int t = threadIdx.x;        // 0..127
  int f = t >> 4;             // 0..7  (0-3 low, 4-7 high)
  int j = t & 15;             // output tap
  int n = f & 3;
  const float* src = (f < 4) ? (lowf + n * 16) : (highf + n * 16);
  double center = ((double)n + 0.5) / 4.0;
  double target = (f < 4) ? center * 0.5 : center * 0.5 + 0.5;

  float fin[16];
#pragma unroll
  for (int m = 0; m < 16; ++m) fin[m] = src[m];

  double acc = 0.0;
  for (int k = 0; k <= 32; ++k) {
    double re = 0.0, im = 0.0;
    for (int m = 0; m < 16; ++m) {
      double ang = -2.0 * M_PI * (double)(k * m) / 64.0;
      re += (double)fin[m] * cos(ang);
      im += (double)fin[m] * sin(ang);
    }
    double fr = (double)k / 32.0;
    double d = (fr - target) * 4.0;            // width = 1/4
    double wk = exp(-d * d);
    double ck = (k == 0 || k == 32) ? 1.0 : 2.0;
    double ang2 = 2.0 * M_PI * (double)(k * j) / 64.0;
    acc += ck * wk * (re * cos(ang2) - im * sin(ang2));
  }
  float val = (float)(acc / 64.0);

  __shared__ float sq[128];
  sq[t] = val * val;
  __syncthreads();
  float ss = 0.f;
  int base = f << 4;
#pragma unroll
  for (int m = 0; m < 16; ++m) ss += sq[base + m];
  float nrm = fmaxf(sqrtf(ss), 1e-12f);
  filt_out[t] = val / nrm * 1.41421356237309504880f;
}

// ---------------------------------------------------------------------------
// Phase 1b: build banded-Toeplitz A fragments in the exact per-lane VGPR
// layout V_WMMA_F32_16X16X4_F32 expects:
//   lanes 0-15 -> Klocal {0,1}, lanes 16-31 -> Klocal {2,3};  .x/.y = K pair
//   A[i][k] = filt[k-2i] if 0 <= k-2i < 16 else 0, k = 4q + Klocal
// Table: 4 channels x 2 filters x 12 K-chunks x 32 lanes x float2 = 24 KB.
// ---------------------------------------------------------------------------
__global__ void build_toeplitz_kernel(const float* __restrict__ filt,
                                      float* __restrict__ toep) {
  int idx = blockIdx.x * blockDim.x + threadIdx.x;   // 0..3071
  if (idx >= 4 * 2 * 12 * 32) return;
  int lane = idx & 31;
  int chunk = idx >> 5;          // 0..95 = ((n*2+s)*12+q)
  int q = chunk % 12;
  int s = (chunk / 12) & 1;
  int n = chunk / 24;
  int i = lane & 15;
  int kh = lane >> 4;
  int t0 = 4 * q + 2 * kh - 2 * i;
  const float* fsrc = filt + (s ? (4 + n) : n) * 16;
  int t0c = t0 < 0 ? 0 : (t0 > 15 ? 15 : t0);
  int t1 = t0 + 1;
  int t1c = t1 < 0 ? 0 : (t1 > 15 ? 15 : t1);
  float ax = (t0 == t0c) ? fsrc[t0c] : 0.f;
  float ay = (t1 == t1c) ? fsrc[t1c] : 0.f;
  toep[idx * 2 + 0] = ax;
  toep[idx * 2 + 1] = ay;
}

// ---------------------------------------------------------------------------
// Phase 2: one DWT level. Stride-2 16-tap correlation as banded-Toeplitz GEMM
// on V_WMMA_F32_16X16X4_F32 (full fp32 => matches reference numerics).
//   D[i][j] = sum_k A[i][k]*B[k][j], k in [0,48)
//   B[k][j] = x_row_j[2*o0 - 15 + k]  (zero-padded)
//   => D[i][j] = sum_t filt[t] * x_j[2*(o0+i)+t-15]
// One wave = 16 outputs x 16 batch rows for one wavelet channel; 12 chained
// WMMAs per filter, B shared between low-pass and high-pass chains.
// Interior tiles: unpredicated loads (overlappable). Edge tiles: branchless
// clamp+select. Tile guard is wave-uniform so EXEC is all-ones at every WMMA.
// ---------------------------------------------------------------------------
__global__ __launch_bounds__(32 * WPB) void dwt_level_kernel(
    const float* __restrict__ xin, long long sB, long long sN,
    int Lin, int Lout,
    float* __restrict__ aout, long long aStr,
    float* __restrict__ dout, long long dStr,
    const float* __restrict__ toep) {
  const int lane = threadIdx.x & 31;
  const int wv = threadIdx.x >> 5;
  const int o0 = (blockIdx.x * WPB + wv) * 16;
  if (o0 >= Lout) return;                 // wave-uniform

  const int g = blockIdx.y;               // 16 row-groups: 4 wavelets x 4 batch-quads
  const int n = g & 3;                    // wavelet channel
  const int gb = g >> 2;                  // batch quad
  const int i = lane & 15;
  const int kh = lane >> 4;
  const int b_idx = gb * 16 + i;

  // A fragments: coalesced b64 loads from the precomputed table.
  const v2f* tlo = (const v2f*)(toep) + (size_t)(n * 2 + 0) * 12 * 32 + lane;
  const v2f* thi = (const v2f*)(toep) + (size_t)(n * 2 + 1) * 12 * 32 + lane;
  v2f alo[12], ahi[12];
#pragma unroll
  for (int q = 0; q < 12; ++q) {
    alo[q] = tlo[q * 32];
    ahi[q] = thi[q * 32];
  }

  const float* xrow = xin + (long long)b_idx * sB + (long long)n * sN;
  const int pbase = 2 * o0 - 15 + 2 * kh;

  // Gather all 24 B scalars up front so loads pipeline ahead of the WMMAs.
  float bxv[24];
  const bool interior = (2 * o0 - 15 >= 0) && (2 * o0 + 32 < Lin); // wave-uniform
  if (interior) {
#pragma unroll
    for (int q = 0; q < 12; ++q) {
      int p = pbase + 4 * q;
      bxv[2 * q + 0] = xrow[p];
      bxv[2 * q + 1] = xrow[p + 1];
    }
  } else {
#pragma unroll
    for (int q = 0; q < 12; ++q) {
      int p = pbase + 4 * q;
      int pc = p < 0 ? 0 : (p >= Lin ? Lin - 1 : p);
      float v0 = xrow[pc];
      int p1 = p + 1;
      int p1c = p1 < 0 ? 0 : (p1 >= Lin ? Lin - 1 : p1);
      float v1 = xrow[p1c];
      bxv[2 * q + 0] = (p == pc) ? v0 : 0.f;
      bxv[2 * q + 1] = (p1 == p1c) ? v1 : 0.f;
    }
  }

  v8f ca = {};   // low-pass (approx) accumulator
  v8f cd = {};   // high-pass (detail) accumulator
#pragma unroll
  for (int q = 0; q < 12; ++q) {
    v2f bb; bb.x = bxv[2 * q]; bb.y = bxv[2 * q + 1];
    ca = __builtin_amdgcn_wmma_f32_16x16x4_f32(false, alo[q], false, bb,
                                               (short)0, ca, false, false);
    cd = __builtin_amdgcn_wmma_f32_16x16x4_f32(false, ahi[q], false, bb,
                                               (short)0, cd, false, false);
  }

  // D layout: VGPR r holds rows M=r (lanes0-15) and M=r+8 (lanes16-31), N=lane%16.
  float* ar = aout + (long long)(b_idx * 4 + n) * aStr;
  float* dr = dout + (long long)(b_idx * 4 + n) * dStr;
#pragma unroll
  for (int r = 0; r < 8; ++r) {
    int o = o0 + r + 8 * kh;
    if (o < Lout) {
      ar[o] = ca[r];
      dr[o] = cd[r];
    }
  }
}

// ---------------------------------------------------------------------------
// Host driver: filter kernel -> Toeplitz table -> 6 level kernels with
// ping-pong approx buffers in d_ws; details (and final approx) stream into
// d_out at their concatenation offsets: [approx5 | d5 | d4 | d3 | d2 | d1 | d0]
// ---------------------------------------------------------------------------
extern "C" void kernel_launch(void* const* d_in, const int* in_sizes, int n_in,
                              void* d_out, int out_size, void* d_ws, size_t ws_size,
                              hipStream_t stream) {
  const float* signal = (const float*)d_in[0];   // [64, 131072]
  const float* lowf   = (const float*)d_in[1];   // [4, 16]
  const float* highf  = (const float*)d_in[2];   // [4, 16]
  float* out = (float*)d_out;                    // [64, 4, 131166]
  float* ws  = (float*)d_ws;

  float* filt = ws;
  float* toep = ws + WS_TOEP;
  float* bufA = ws + WS_BUF;                     // 256 rows x 65544
  float* bufB = bufA + (size_t)256 * 65544;      // 256 rows x 32780
  // total ws use: ~100.7 MB

  hipLaunchKernelGGL(constrain_filters_kernel, dim3(1), dim3(128), 0, stream,
                     lowf, highf, filt);
  hipLaunchKernelGGL(build_toeplitz_kernel, dim3(12), dim3(256), 0, stream,
                     filt, toep);

  const long long T = T_OUT;
  const int dof[LEVELS] = {65622, 32842, 16444, 8237, 4126, 2063};

  const float* xin = signal;
  long long sB = SIG_LEN, sN = 0;   // level 0: broadcast over wavelet dim
  int L = SIG_LEN;
  for (int lev = 0; lev < LEVELS; ++lev) {
    int Lout = (L + 14) / 2 + 1;    // floor((L + 30 - 16)/2) + 1
    float* aout;
    long long aStr;
    if (lev == LEVELS - 1) { aout = out; aStr = T; }
    else if ((lev & 1) == 0) { aout = bufA; aStr = Lout; }
    else                     { aout = bufB; aStr = Lout; }
    float* dout = out + dof[lev];

    int tiles = (Lout + 15) / 16;
    dim3 grid((tiles + WPB - 1) / WPB, 16);
    hipLaunchKernelGGL(dwt_level_kernel, grid, dim3(32 * WPB), 0, stream,
                       xin, sB, sN, L, Lout, aout, aStr, dout, T, toep);

    xin = aout;
    sB = 4LL * Lout;
    sN = Lout;
    L = Lout;
  }
}